// SCRAPLLoss_14577119002639
// MI455X (gfx1250) — compile-verified
//
#include <hip/hip_runtime.h>
#include <cstdint>
#include <cstddef>

// ---------------------------------------------------------------------------
// SCRAPL single-path scattering loss for gfx1250 (MI455X).
//  * time FFTs (32768-pt) done fully inside 256KB LDS, one FFT chain / WGP
//  * filter-axis fft/diag/ifft collapsed to a 96x96 complex circulant matmul
//    executed with V_WMMA_F32_16X16X4_F32 (fp32 tensor path, no precision loss)
//  * Y stored as de-interleaved re/im planes, layout [b][t][g] so WMMA
//    operands are direct aligned b64 loads (no swizzle movs in the hot loop)
//  * B-tiles staged with global_load_async_to_lds_b128 + s_wait_asynccnt
// ---------------------------------------------------------------------------

#define NFFT   32768
#define LOG2N  15
#define N1F    96      // J*Q1
#define LFR    128
#define WIN    4096
#define NWIN   8       // NFFT / WIN
#define PI_F   3.14159265358979323846f

typedef float v2f __attribute__((ext_vector_type(2)));
typedef float v8f __attribute__((ext_vector_type(8)));

// ---------------- workspace layout (float offsets) --------------------------
#define OFF_AR    ((size_t)0)                       // 96*96
#define OFF_AI    (OFF_AR  + (size_t)(96*96))       // 96*96
#define OFF_AIN   (OFF_AI  + (size_t)(96*96))       // 96*96 (negated Ai)
#define OFF_SX    (OFF_AIN + (size_t)(96*96))       // 16*96*8
#define OFF_SXT   (OFF_SX  + (size_t)(16*96*8))     // 16*96*8
#define OFF_XHAT  (OFF_SXT + (size_t)(16*96*8))     // 16*NFFT*2 (float2)
#define OFF_YR    (OFF_XHAT + (size_t)16*NFFT*2)    // 16*NFFT*96  (re plane)
#define OFF_YI    (OFF_YR  + (size_t)16*NFFT*N1F)   // 16*NFFT*96  (im plane)

// ---------------------------------------------------------------------------
// In-LDS radix-2 FFT helpers. DIF: natural -> bit-reversed.  DIT inverse:
// bit-reversed -> natural.  Matched pair => no explicit reordering needed.
// ---------------------------------------------------------------------------
__device__ __forceinline__ void fft_dif_fwd(float2* sh, int tid, int T) {
  for (int stage = 0; stage < LOG2N; ++stage) {
    const int hb   = (LOG2N - 1) - stage;   // log2(half)
    const int half = 1 << hb;
    __syncthreads();
    for (int j = tid; j < (NFFT / 2); j += T) {
      const int blk = j >> hb;
      const int k   = j & (half - 1);
      const int i0  = (blk << (hb + 1)) + k;
      const int i1  = i0 + half;
      float2 a = sh[i0], b = sh[i1];
      float2 s = make_float2(a.x + b.x, a.y + b.y);
      float2 d = make_float2(a.x - b.x, a.y - b.y);
      const float ang = -2.0f * PI_F * (float)k / (float)(half << 1);
      float sn, cs; __sincosf(ang, &sn, &cs);
      sh[i0] = s;
      sh[i1] = make_float2(d.x * cs - d.y * sn, d.x * sn + d.y * cs);
    }
  }
  __syncthreads();
}

__device__ __forceinline__ void fft_dit_inv(float2* sh, int tid, int T) {
  for (int stage = 0; stage < LOG2N; ++stage) {
    const int half = 1 << stage;
    __syncthreads();
    for (int j = tid; j < (NFFT / 2); j += T) {
      const int blk = j >> stage;
      const int k   = j & (half - 1);
      const int i0  = (blk << (stage + 1)) + k;
      const int i1  = i0 + half;
      const float ang = 2.0f * PI_F * (float)k / (float)(half << 1);
      float sn, cs; __sincosf(ang, &sn, &cs);
      float2 a = sh[i0], b = sh[i1];
      float2 bw = make_float2(b.x * cs - b.y * sn, b.x * sn + b.y * cs);
      sh[i0] = make_float2(a.x + bw.x, a.y + bw.y);
      sh[i1] = make_float2(a.x - bw.x, a.y - bw.y);
    }
  }
  __syncthreads();
}

__device__ __forceinline__ int brev15(int p) {
  return (int)(__brev((unsigned)p) >> 17);
}

// ---------------------------------------------------------------------------
// K0: build circulant operator  A[m,n] = c[(m-n) & 127],  c = ifft(psi_fr_sel)
// ---------------------------------------------------------------------------
__global__ void __launch_bounds__(128) k_build_A(const float* __restrict__ psi_fr,
                                                 const int* __restrict__ nfrp,
                                                 float* __restrict__ ws) {
  __shared__ float cr[LFR], ci[LFR];
  const float* w = psi_fr + (size_t)(*nfrp) * LFR;
  const int tid = threadIdx.x;
  if (tid < LFR) {
    float sr = 0.f, si = 0.f;
    for (int k = 0; k < LFR; ++k) {
      const float ang = 2.0f * PI_F * (float)(k * tid) / (float)LFR;
      float sn, cs; __sincosf(ang, &sn, &cs);
      sr += w[k] * cs;
      si += w[k] * sn;
    }
    cr[tid] = sr * (1.0f / LFR);
    ci[tid] = si * (1.0f / LFR);
  }
  __syncthreads();
  float* Ar  = ws + OFF_AR;
  float* Ai  = ws + OFF_AI;
  float* Ain = ws + OFF_AIN;
  for (int idx = tid; idx < N1F * N1F; idx += blockDim.x) {
    const int m = idx / N1F;
    const int n = idx - m * N1F;
    const int d = (m - n) & (LFR - 1);
    Ar[idx]  =  cr[d];
    Ai[idx]  =  ci[d];
    Ain[idx] = -ci[d];
  }
}

// ---------------------------------------------------------------------------
// zero the two pooled-output buffers (atomics accumulate into them)
// ---------------------------------------------------------------------------
__global__ void k_zero(float* __restrict__ p, int n) {
  int i = blockIdx.x * blockDim.x + threadIdx.x;
  if (i < n) p[i] = 0.0f;
}

// ---------------------------------------------------------------------------
// K1: forward FFT of the input signal, one batch per workgroup, whole FFT
// in 256KB LDS.  Output stored in bit-reversed order.
// ---------------------------------------------------------------------------
__global__ void __launch_bounds__(1024) k_fft_fwd(const float* __restrict__ x,
                                                  float2* __restrict__ Xhat) {
  extern __shared__ float2 sh[];
  const int b = blockIdx.x;
  const int tid = threadIdx.x, T = blockDim.x;
  for (int i = tid; i < NFFT; i += T)
    sh[i] = make_float2(x[(size_t)b * NFFT + i], 0.0f);
  fft_dif_fwd(sh, tid, T);
  float2* out = Xhat + (size_t)b * NFFT;
  for (int i = tid; i < NFFT; i += T) out[i] = sh[i];
}

// ---------------------------------------------------------------------------
// K2: fused per-(batch,filter) path:
//   X*psi1 -> ifft -> |.| -> fft -> *psi2 -> ifft  -> Y (re/im planes)
// all intermediate FFT state lives in LDS (256KB = 32768 complex f32).
// Output layout: Yr/Yi[b][t][g]  (filter axis contiguous for the WMMA stage)
// ---------------------------------------------------------------------------
__global__ void __launch_bounds__(1024) k_path(const float2* __restrict__ Xhat,
                                               const float* __restrict__ psi1,
                                               const float* __restrict__ psi2,
                                               const int* __restrict__ n2p,
                                               float* __restrict__ Yr,
                                               float* __restrict__ Yi) {
  extern __shared__ float2 sh[];
  const int b = blockIdx.x / N1F;
  const int f = blockIdx.x % N1F;
  const int tid = threadIdx.x, T = blockDim.x;
  const float*  p1 = psi1 + (size_t)f * NFFT;
  const float*  p2 = psi2 + (size_t)(*n2p) * NFFT;
  const float2* Xb = Xhat + (size_t)b * NFFT;
  const float invN = 1.0f / (float)NFFT;

  // multiply by psi1 (data is bit-reversed; filter indexed by brev)
  for (int p = tid; p < NFFT; p += T) {
    const float w = p1[brev15(p)];
    float2 v = Xb[p];
    sh[p] = make_float2(v.x * w, v.y * w);
  }
  fft_dit_inv(sh, tid, T);                       // -> U1 (needs 1/N)

  for (int p = tid; p < NFFT; p += T) {
    float2 v = sh[p];
    sh[p] = make_float2(sqrtf(v.x * v.x + v.y * v.y) * invN, 0.0f);
  }
  fft_dif_fwd(sh, tid, T);                       // fft(U1), bit-reversed

  for (int p = tid; p < NFFT; p += T) {
    const float w = p2[brev15(p)];
    float2 v = sh[p];
    sh[p] = make_float2(v.x * w, v.y * w);
  }
  fft_dit_inv(sh, tid, T);                       // -> Y (needs 1/N)

  // de-interleaved, filter-contiguous store: Yr/Yi[(b*NFFT + t)*96 + f]
  float* YrP = Yr + (size_t)b * NFFT * N1F + f;
  float* YiP = Yi + (size_t)b * NFFT * N1F + f;
  for (int p = tid; p < NFFT; p += T) {
    float2 v = sh[p];
    YrP[(size_t)p * N1F] = v.x * invN;
    YiP[(size_t)p * N1F] = v.y * invN;
  }
}

// ---------------------------------------------------------------------------
// K3: frequency-axis circulant matmul via WMMA fp32, fused with |.| and
// window pooling.  Block = 192 threads (6 waves); each wave owns one 16-row
// filter tile; the whole block shares one 96x16 Y tile (re+im planes, 12 KB)
// staged into LDS with async-to-LDS b128 copies.
// ---------------------------------------------------------------------------
__global__ void __launch_bounds__(192) k_freq_wmma(const float* __restrict__ Yr,
                                                   const float* __restrict__ Yi,
                                                   const float* __restrict__ ws,
                                                   float* __restrict__ S) {
  // re plane: Bt[0 .. 16*96), im plane: Bt[16*96 .. 2*16*96); index n*96+g
  __shared__ __align__(16) float Bt[2 * 16 * N1F];

  const int ttile = blockIdx.x;                  // 0..2047
  const int b     = blockIdx.y;                  // 0..15
  const int t0    = ttile * 16;
  const int tid   = threadIdx.x;
  const int wave  = tid >> 5;                    // 0..5  -> filter tile
  const int lane  = tid & 31;

  // ---- async stage of Yr/Yi[b, t0..t0+15, 0..95] into LDS ----------------
  // each t-row is 96 floats = 384B = 24 x 16B chunks; 2 planes x 16 rows.
  {
    const unsigned lds_base = (unsigned)(size_t)(&Bt[0]);
    const char* gR = (const char*)(Yr + ((size_t)b * NFFT + t0) * N1F);
    const char* gI = (const char*)(Yi + ((size_t)b * NFFT + t0) * N1F);
    for (int q = tid; q < 2 * 16 * 24; q += 192) {
      const int plane = q / 384;
      const int c     = q - plane * 384;
      const int n     = c / 24;
      const int sub   = c - n * 24;
      const char* gb  = (plane ? gI : gR) + (size_t)n * (N1F * 4) + (size_t)sub * 16;
      const unsigned long long gaddr = (unsigned long long)gb;
      const unsigned lds = lds_base + (unsigned)(plane * (16 * N1F * 4) + n * 384 + sub * 16);
      asm volatile("global_load_async_to_lds_b128 %0, %1, off"
                   :: "v"(lds), "v"(gaddr) : "memory");
    }
    asm volatile("s_wait_asynccnt 0" ::: "memory");
    __syncthreads();
  }

  const float* Ar  = ws + OFF_AR;
  const float* Ai  = ws + OFF_AI;
  const float* Ain = ws + OFF_AIN;
  const float* BtR = &Bt[0];
  const float* BtI = &Bt[16 * N1F];

  const int f0  = wave * 16;
  const int m   = f0 + (lane & 15);              // A-matrix row for this lane
  const int ksb = (lane >> 4) * 2;               // K sub-pair {0,1} or {2,3}
  const int n   = lane & 15;                     // B/C column

  v8f Dr = {};
  v8f Di = {};

  for (int kc = 0; kc < N1F / 4; ++kc) {         // 24 K-chunks of 4
    const int g0 = kc * 4 + ksb;                 // even -> 8B aligned pairs

    // A operands: adjacent in g -> single aligned b64 loads
    const v2f ar = *(const v2f*)(Ar  + (size_t)m * N1F + g0);
    const v2f ai = *(const v2f*)(Ai  + (size_t)m * N1F + g0);
    const v2f an = *(const v2f*)(Ain + (size_t)m * N1F + g0);

    // B operands: adjacent in g in LDS -> single aligned ds_load_b64 each
    const v2f br = *(const v2f*)(BtR + n * N1F + g0);
    const v2f bi = *(const v2f*)(BtI + n * N1F + g0);

    // complex accumulate: Dr += Ar*Yr - Ai*Yi ; Di += Ar*Yi + Ai*Yr
    Dr = __builtin_amdgcn_wmma_f32_16x16x4_f32(false, ar, false, br,
                                               (short)0, Dr, false, false);
    Dr = __builtin_amdgcn_wmma_f32_16x16x4_f32(false, an, false, bi,
                                               (short)0, Dr, false, false);
    Di = __builtin_amdgcn_wmma_f32_16x16x4_f32(false, ar, false, bi,
                                               (short)0, Di, false, false);
    Di = __builtin_amdgcn_wmma_f32_16x16x4_f32(false, ai, false, br,
                                               (short)0, Di, false, false);
  }

  // ---- |.| + pool: sum over the 16 t's of this tile, atomically add into S
  const int widx  = t0 / WIN;                    // all 16 t's share one window
  const float wsc = 1.0f / (float)WIN;
  for (int r = 0; r < 8; ++r) {
    float mag = sqrtf(Dr[r] * Dr[r] + Di[r] * Di[r]);
    // reduce across the 16 lanes of each half-wave (same m, all n)
    mag += __shfl_xor(mag, 1, 32);
    mag += __shfl_xor(mag, 2, 32);
    mag += __shfl_xor(mag, 4, 32);
    mag += __shfl_xor(mag, 8, 32);
    if ((lane & 15) == 0) {
      const int mm = f0 + r + ((lane >> 4) ? 8 : 0);
      atomicAdd(&S[((size_t)b * N1F + mm) * NWIN + widx], mag * wsc);
    }
  }
}

// ---------------------------------------------------------------------------
// K4: distance reduction:  mean_b sqrt( sum_{f,w} (Sxt - Sx)^2 )
// ---------------------------------------------------------------------------
__global__ void __launch_bounds__(256) k_dist(const float* __restrict__ Sx,
                                              const float* __restrict__ Sxt,
                                              float* __restrict__ out) {
  __shared__ float red[256];
  const int tid = threadIdx.x;
  float acc = 0.0f;
  for (int b = 0; b < 16; ++b) {
    float s = 0.0f;
    for (int i = tid; i < N1F * NWIN; i += 256) {
      const float d = Sxt[(size_t)b * N1F * NWIN + i] - Sx[(size_t)b * N1F * NWIN + i];
      s += d * d;
    }
    red[tid] = s;
    __syncthreads();
    for (int st = 128; st > 0; st >>= 1) {
      if (tid < st) red[tid] += red[tid + st];
      __syncthreads();
    }
    if (tid == 0) acc += sqrtf(red[0]);
    __syncthreads();
  }
  if (tid == 0) out[0] = acc * (1.0f / 16.0f);
}

// ---------------------------------------------------------------------------
extern "C" void kernel_launch(void* const* d_in, const int* in_sizes, int n_in,
                              void* d_out, int out_size, void* d_ws, size_t ws_size,
                              hipStream_t stream) {
  (void)in_sizes; (void)n_in; (void)out_size; (void)ws_size;

  const float* x     = (const float*)d_in[0];
  const float* xt    = (const float*)d_in[1];
  const int*   n2p   = (const int*)d_in[2];
  const int*   nfrp  = (const int*)d_in[3];
  const float* psi1  = (const float*)d_in[4];
  const float* psi2  = (const float*)d_in[5];
  const float* psifr = (const float*)d_in[6];
  float* out = (float*)d_out;

  float*  wsf  = (float*)d_ws;
  float*  Sx   = wsf + OFF_SX;
  float*  Sxt  = wsf + OFF_SXT;
  float2* Xhat = (float2*)(wsf + OFF_XHAT);
  float*  Yr   = wsf + OFF_YR;
  float*  Yi   = wsf + OFF_YI;

  const size_t ldsBytes = (size_t)NFFT * sizeof(float2);   // 256 KB

  // operator + clear pooled outputs
  hipLaunchKernelGGL(k_build_A, dim3(1), dim3(128), 0, stream, psifr, nfrp, wsf);
  hipLaunchKernelGGL(k_zero, dim3((2 * 16 * N1F * NWIN + 255) / 256), dim3(256),
                     0, stream, Sx, 2 * 16 * N1F * NWIN);

  // ---- signal 1: x ----
  hipLaunchKernelGGL(k_fft_fwd, dim3(16), dim3(1024), ldsBytes, stream, x, Xhat);
  hipLaunchKernelGGL(k_path, dim3(16 * N1F), dim3(1024), ldsBytes, stream,
                     Xhat, psi1, psi2, n2p, Yr, Yi);
  hipLaunchKernelGGL(k_freq_wmma, dim3(NFFT / 16, 16), dim3(192), 0, stream,
                     Yr, Yi, wsf, Sx);

  // ---- signal 2: x_target (reuse Xhat / Y buffers) ----
  hipLaunchKernelGGL(k_fft_fwd, dim3(16), dim3(1024), ldsBytes, stream, xt, Xhat);
  hipLaunchKernelGGL(k_path, dim3(16 * N1F), dim3(1024), ldsBytes, stream,
                     Xhat, psi1, psi2, n2p, Yr, Yi);
  hipLaunchKernelGGL(k_freq_wmma, dim3(NFFT / 16, 16), dim3(192), 0, stream,
                     Yr, Yi, wsf, Sxt);

  // ---- distance ----
  hipLaunchKernelGGL(k_dist, dim3(1), dim3(256), 0, stream, Sx, Sxt, out);
}